// YOLOXv2_69552700391734
// MI455X (gfx1250) — compile-verified
//
#include <hip/hip_runtime.h>
#include <math.h>

#ifndef __has_builtin
#define __has_builtin(x) 0
#endif

#if __has_builtin(__builtin_amdgcn_global_load_async_to_lds_b128) && \
    __has_builtin(__builtin_amdgcn_s_wait_asynccnt)
#define USE_ASYNC_LDS 1
#else
#define USE_ASYNC_LDS 0
#endif

#define NUM_CLASSES 80
#define TOPK        1000
#define NLEVEL      3
#define TOTAL       (NLEVEL * TOPK)   // 3000
#define SORTN       4096
#define CONF_LOGIT  (-2.9444389791664403f)  // log(0.05/0.95)
#define NMS_THR     0.6f
#define CLASS_OFF   10000.0f

#if USE_ASYNC_LDS
typedef int v4i_t __attribute__((ext_vector_type(4)));
typedef __attribute__((address_space(1))) v4i_t* gv4i_p;   // global
typedef __attribute__((address_space(3))) v4i_t* lv4i_p;   // LDS
#endif

// ---------- order-preserving float<->uint key ----------
__device__ __forceinline__ unsigned tokey(float f) {
    unsigned u = __float_as_uint(f);
    return (u & 0x80000000u) ? ~u : (u | 0x80000000u);
}
__device__ __forceinline__ float fromkey(unsigned k) {
    unsigned u = (k & 0x80000000u) ? (k & 0x7fffffffu) : ~k;
    return __uint_as_float(u);
}

// ======================================================================
// Histogram pass (phases 0/1/2 of radix select).
// Double-buffered async-to-LDS pipeline: load chunk c+1 overlaps the LDS
// histogram of chunk c. ASYNCcnt completes in order, so s_wait_asynccnt(1)
// after issuing the prefetch guarantees the current chunk is resident.
// Addresses are clamped (not guarded) so every wave issues exactly one
// async op per iteration -> counter arithmetic stays exact.
// ======================================================================
__global__ __launch_bounds__(256) void hist_pass(const float4* __restrict__ data, int n4,
                                                 unsigned* __restrict__ ghist,
                                                 const unsigned* __restrict__ sel,
                                                 int phase, int nbins)
{
    __shared__ unsigned lhist[4096];
    __shared__ __align__(16) float4 tile[2][256];
    const int tid = threadIdx.x;
    for (int i = tid; i < nbins; i += 256) lhist[i] = 0u;
    const unsigned B1 = sel[0];
    const unsigned P2 = (sel[0] << 12) | sel[2];
    __syncthreads();

    const int step = gridDim.x * 256;
    int base = blockIdx.x * 256;
    int buf = 0;

#if USE_ASYNC_LDS
    if (base < n4) {   // block-uniform prologue: stage first chunk
        int cidx = base + tid; if (cidx >= n4) cidx = n4 - 1;
        __builtin_amdgcn_global_load_async_to_lds_b128(
            (gv4i_p)(const_cast<float4*>(data) + cidx), (lv4i_p)(&tile[0][tid]), 0, 0);
    }
#endif

    for (; base < n4; base += step) {
        const int idx = base + tid;
#if USE_ASYNC_LDS
        const int nextBase = base + step;         // block-uniform
        if (nextBase < n4) {
            int cidx = nextBase + tid; if (cidx >= n4) cidx = n4 - 1;
            __builtin_amdgcn_global_load_async_to_lds_b128(
                (gv4i_p)(const_cast<float4*>(data) + cidx),
                (lv4i_p)(&tile[buf ^ 1][tid]), 0, 0);
            __builtin_amdgcn_s_wait_asynccnt(1);  // oldest (current chunk) done
        } else {
            __builtin_amdgcn_s_wait_asynccnt(0);  // drain
        }
#else
        if (idx < n4) tile[buf][tid] = data[idx];
#endif
        __syncthreads();                          // current chunk visible to all waves
        if (idx < n4) {
            const float4 v = tile[buf][tid];
            const float e[4] = {v.x, v.y, v.z, v.w};
#pragma unroll
            for (int q = 0; q < 4; ++q) {
                const unsigned key = tokey(e[q]);
                if (phase == 0) {
                    atomicAdd(&lhist[key >> 20], 1u);
                } else if (phase == 1) {
                    if ((key >> 20) == B1) atomicAdd(&lhist[(key >> 8) & 0xFFFu], 1u);
                } else {
                    if ((key >> 8) == P2) atomicAdd(&lhist[key & 0xFFu], 1u);
                }
            }
        }
        __syncthreads();                          // done reading before buffer reuse
        buf ^= 1;
    }
    for (int i = tid; i < nbins; i += 256) {
        const unsigned c = lhist[i];
        if (c) atomicAdd(&ghist[i], c);
    }
}

// ======================================================================
// Suffix-scan over histogram bins; find k-th-largest bucket. grid.x = level
// sel layout per level (16 u32):
// [0]=B1 [1]=cGT1 [2]=B2 [3]=cGT2 [4]=B3 [5]=cGT3 [6]=T [7]=cGTtot
// [8]=needTie [9]=ctrG [10]=ctrT
// ======================================================================
__global__ __launch_bounds__(1024) void scan_pass(const unsigned* __restrict__ histBase, int nbins,
                                                  unsigned* __restrict__ selBase, int phase, int K)
{
    __shared__ unsigned A[4096], B[4096];
    const int lvl = blockIdx.x;
    const unsigned* hist = histBase + (size_t)lvl * nbins;
    unsigned* sel = selBase + lvl * 16;

    for (int i = threadIdx.x; i < nbins; i += 1024) A[i] = hist[i];
    __syncthreads();

    int pp = 0;
    for (int off = 1; off < nbins; off <<= 1) {
        unsigned* src = pp ? B : A;
        unsigned* dst = pp ? A : B;
        for (int i = threadIdx.x; i < nbins; i += 1024) {
            unsigned v = src[i];
            if (i + off < nbins) v += src[i + off];
            dst[i] = v;
        }
        __syncthreads();
        pp ^= 1;
    }
    unsigned* suf = pp ? B : A;  // suf[i] = count(bin >= i)

    unsigned k;
    if (phase == 0)      k = (unsigned)K;
    else if (phase == 1) k = (unsigned)K - sel[1];
    else                 k = (unsigned)K - sel[1] - sel[3];

    for (int i = threadIdx.x; i < nbins; i += 1024) {
        const unsigned s0 = suf[i];
        const unsigned s1 = (i + 1 < nbins) ? suf[i + 1] : 0u;
        if (s0 >= k && s1 < k) {       // unique crossing bin
            if (phase == 0)      { sel[0] = (unsigned)i; sel[1] = s1; }
            else if (phase == 1) { sel[2] = (unsigned)i; sel[3] = s1; }
            else {
                const unsigned T   = (sel[0] << 20) | (sel[2] << 8) | (unsigned)i;
                const unsigned tot = sel[1] + sel[3] + s1;  // count(key > T)
                sel[4] = (unsigned)i; sel[5] = s1;
                sel[6] = T; sel[7] = tot; sel[8] = (unsigned)K - tot;
            }
        }
    }
}

// ======================================================================
// Compact: gather top-K (key,flatIdx) per level
// ======================================================================
__global__ __launch_bounds__(256) void compact_pass(const float* __restrict__ data, int n,
                                                    unsigned* sel, uint2* __restrict__ cand)
{
    const unsigned T = sel[6], base = sel[7], needTie = sel[8];
    for (int i = blockIdx.x * 256 + threadIdx.x; i < n; i += gridDim.x * 256) {
        const unsigned key = tokey(data[i]);
        if (key > T) {
            const unsigned p = atomicAdd(&sel[9], 1u);
            if (p < (unsigned)TOPK) cand[p] = make_uint2(key, (unsigned)i);
        } else if (key == T) {
            const unsigned t = atomicAdd(&sel[10], 1u);
            if (t < needTie) {
                const unsigned p = base + t;
                if (p < (unsigned)TOPK) cand[p] = make_uint2(key, (unsigned)i);
            }
        }
    }
}

// ======================================================================
// Finalize: sort 3000 by score desc, decode boxes, greedy class-aware NMS,
// write outputs. Single workgroup, 1024 threads (32 waves), LDS phase-reused.
// ======================================================================
__global__ __launch_bounds__(1024) void finalize_kernel(const uint2* __restrict__ cand,
                                                        const float* __restrict__ reg0,
                                                        const float* __restrict__ reg1,
                                                        const float* __restrict__ reg2,
                                                        float* __restrict__ out)
{
    __shared__ __align__(16) unsigned char smem[49152];
    float* negkey = (float*)smem;            // [4096] phase A
    int*   pay    = (int*)(smem + 16384);    // [4096] phase A
    const int tid = threadIdx.x;
    const float* regs[3] = {reg0, reg1, reg2};

    // ---- load candidates, build sort keys (neg so ascending == score desc) ----
    for (int s = 0; s < 4; ++s) {
        const int pos = tid + s * 1024;
        if (pos < TOTAL) {
            const uint2 cd = cand[pos];
            const float logit = fromkey(cd.x);
            const bool valid = (logit > CONF_LOGIT);         // NaN-safe
            const float sig = 1.0f / (1.0f + expf(-logit));
            const float sc = valid ? sig : -1.0f;            // matches where(valid, s, -1)
            negkey[pos] = -sc;
            pay[pos] = pos;
        } else { negkey[pos] = 3.0e38f; pay[pos] = -1; }
    }
    __syncthreads();

    // ---- bitonic sort ascending on negkey ----
    for (int k = 2; k <= SORTN; k <<= 1) {
        for (int j = k >> 1; j > 0; j >>= 1) {
            for (int t = tid; t < SORTN; t += 1024) {
                const int ixj = t ^ j;
                if (ixj > t) {
                    const float av = negkey[t], bv = negkey[ixj];
                    const bool up = ((t & k) == 0);
                    if ((av > bv) == up) {
                        negkey[t] = bv; negkey[ixj] = av;
                        const int tp = pay[t]; pay[t] = pay[ixj]; pay[ixj] = tp;
                    }
                }
            }
            __syncthreads();
        }
    }

    // ---- capture sorted payload, then repurpose LDS for NMS ----
    int myCand[4];
    for (int s = 0; s < 4; ++s) myCand[s] = pay[tid + s * 1024];
    __syncthreads();

    float*    boxes = (float*)smem;                         // [3000*4] offset boxes
    unsigned* keep  = (unsigned*)(smem + TOTAL * 4 * 4);    // 96 words
    if (tid < 96) keep[tid] = 0u;

    float myBox[4][4]; float myScore[4]; int myLabel[4]; bool myValid[4];
    for (int s = 0; s < 4; ++s) {
        const int pos = tid + s * 1024;
        myValid[s] = false; myScore[s] = 0.f; myLabel[s] = -1;
        myBox[s][0] = myBox[s][1] = myBox[s][2] = myBox[s][3] = 0.f;
        if (pos < TOTAL) {
            const int c = myCand[s];
            if (c >= 0) {
                const int level = c / TOPK;
                const uint2 cd = cand[c];
                const float logit = fromkey(cd.x);
                myValid[s] = (logit > CONF_LOGIT);
                myScore[s] = 1.0f / (1.0f + expf(-logit));
                const unsigned flat = cd.y;
                const int a   = (int)(flat / NUM_CLASSES);
                const int cls = (int)(flat % NUM_CLASSES);
                const int w = 320 >> level;
                const float stride = (float)(8 << level);
                const int yy = a / w, xx = a - yy * w;
                const float* r = regs[level] + (size_t)a * 4;
                const float cx = (xx + 0.5f) * stride + r[0] * stride;
                const float cy = (yy + 0.5f) * stride + r[1] * stride;
                const float bw = expf(r[2]) * stride;
                const float bh = expf(r[3]) * stride;
                myBox[s][0] = cx - 0.5f * bw; myBox[s][1] = cy - 0.5f * bh;
                myBox[s][2] = cx + 0.5f * bw; myBox[s][3] = cy + 0.5f * bh;
                myLabel[s] = cls;
                const float off = (float)cls * CLASS_OFF;
                boxes[pos * 4 + 0] = myBox[s][0] + off;
                boxes[pos * 4 + 1] = myBox[s][1] + off;
                boxes[pos * 4 + 2] = myBox[s][2] + off;
                boxes[pos * 4 + 3] = myBox[s][3] + off;
            } else {
                boxes[pos * 4 + 0] = boxes[pos * 4 + 1] =
                boxes[pos * 4 + 2] = boxes[pos * 4 + 3] = 0.f;
            }
        }
    }
    __syncthreads();
    for (int s = 0; s < 4; ++s) {
        const int pos = tid + s * 1024;
        if (pos < TOTAL && myValid[s]) atomicOr(&keep[pos >> 5], 1u << (pos & 31));
    }
    __syncthreads();

    // ---- greedy NMS (keep[i] uniform per block -> barrier skipped on dead rows) ----
    for (int i = 0; i < TOTAL; ++i) {
        const unsigned wv = keep[i >> 5];
        if ((wv >> (i & 31)) & 1u) {
            const float x1 = boxes[i * 4 + 0], y1 = boxes[i * 4 + 1];
            const float x2 = boxes[i * 4 + 2], y2 = boxes[i * 4 + 3];
            const float ai = (x2 - x1) * (y2 - y1);
            for (int j = i + 1 + tid; j < TOTAL; j += 1024) {
                const float bx1 = boxes[j * 4 + 0], by1 = boxes[j * 4 + 1];
                const float bx2 = boxes[j * 4 + 2], by2 = boxes[j * 4 + 3];
                const float xx1 = fmaxf(x1, bx1), yy1 = fmaxf(y1, by1);
                const float xx2 = fminf(x2, bx2), yy2 = fminf(y2, by2);
                const float inter = fmaxf(xx2 - xx1, 0.f) * fmaxf(yy2 - yy1, 0.f);
                const float aj = (bx2 - bx1) * (by2 - by1);
                const float iou = inter / (ai + aj - inter + 1e-9f);
                if (iou > NMS_THR) atomicAnd(&keep[j >> 5], ~(1u << (j & 31)));
            }
            __syncthreads();
        }
    }
    __syncthreads();

    // ---- write outputs: boxes[3000*4] ++ scores[3000] ++ labels[3000] ----
    float* outBoxes  = out;
    float* outScores = out + TOTAL * 4;
    float* outLabels = out + TOTAL * 5;
    for (int s = 0; s < 4; ++s) {
        const int pos = tid + s * 1024;
        if (pos < TOTAL) {
            const bool fin = ((keep[pos >> 5] >> (pos & 31)) & 1u) != 0u;
            outBoxes[pos * 4 + 0] = fin ? myBox[s][0] : 0.f;
            outBoxes[pos * 4 + 1] = fin ? myBox[s][1] : 0.f;
            outBoxes[pos * 4 + 2] = fin ? myBox[s][2] : 0.f;
            outBoxes[pos * 4 + 3] = fin ? myBox[s][3] : 0.f;
            outScores[pos] = fin ? myScore[s] : 0.f;
            outLabels[pos] = fin ? (float)myLabel[s] : -1.0f;
        }
    }
}

// ======================================================================
// Host launcher
// ======================================================================
extern "C" void kernel_launch(void* const* d_in, const int* in_sizes, int n_in,
                              void* d_out, int out_size, void* d_ws, size_t ws_size,
                              hipStream_t stream)
{
    (void)in_sizes; (void)n_in; (void)out_size; (void)ws_size;
    // setup_inputs dict order: cls0, reg0, cls1, reg1, cls2, reg2
    const float* cls[3] = {(const float*)d_in[0], (const float*)d_in[2], (const float*)d_in[4]};
    const float* reg[3] = {(const float*)d_in[1], (const float*)d_in[3], (const float*)d_in[5]};

    unsigned* ws    = (unsigned*)d_ws;
    unsigned* hist1 = ws;                    // 3*4096
    unsigned* hist2 = hist1 + 3 * 4096;      // 3*4096
    unsigned* hist3 = hist2 + 3 * 4096;      // 3*256
    unsigned* sel   = hist3 + 3 * 256;       // 3*16
    uint2*    cand  = (uint2*)(sel + 3 * 16);// 3*1000 uint2 (8B-aligned offset)

    const size_t zeroBytes = (size_t)(3 * 4096 + 3 * 4096 + 3 * 256 + 3 * 16 + 3 * TOPK * 2) * 4;
    (void)hipMemsetAsync(d_ws, 0, zeroBytes, stream);

    const int HW[3] = {320 * 320, 160 * 160, 80 * 80};

    // radix-select pass 0 (top 12 bits)
    for (int l = 0; l < 3; ++l) {
        const int n4 = HW[l] * NUM_CLASSES / 4;
        int blocks = (n4 + 255) / 256; if (blocks > 2048) blocks = 2048;
        hist_pass<<<blocks, 256, 0, stream>>>((const float4*)cls[l], n4,
                                              hist1 + l * 4096, sel + l * 16, 0, 4096);
    }
    scan_pass<<<3, 1024, 0, stream>>>(hist1, 4096, sel, 0, TOPK);

    // pass 1 (mid 12 bits)
    for (int l = 0; l < 3; ++l) {
        const int n4 = HW[l] * NUM_CLASSES / 4;
        int blocks = (n4 + 255) / 256; if (blocks > 2048) blocks = 2048;
        hist_pass<<<blocks, 256, 0, stream>>>((const float4*)cls[l], n4,
                                              hist2 + l * 4096, sel + l * 16, 1, 4096);
    }
    scan_pass<<<3, 1024, 0, stream>>>(hist2, 4096, sel, 1, TOPK);

    // pass 2 (low 8 bits)
    for (int l = 0; l < 3; ++l) {
        const int n4 = HW[l] * NUM_CLASSES / 4;
        int blocks = (n4 + 255) / 256; if (blocks > 2048) blocks = 2048;
        hist_pass<<<blocks, 256, 0, stream>>>((const float4*)cls[l], n4,
                                              hist3 + l * 256, sel + l * 16, 2, 256);
    }
    scan_pass<<<3, 1024, 0, stream>>>(hist3, 256, sel, 2, TOPK);

    // compact exact per-level top-K
    for (int l = 0; l < 3; ++l) {
        const int n = HW[l] * NUM_CLASSES;
        int blocks = (n + 255) / 256; if (blocks > 2048) blocks = 2048;
        compact_pass<<<blocks, 256, 0, stream>>>(cls[l], n, sel + l * 16, cand + l * TOPK);
    }

    // sort + decode + NMS + output
    finalize_kernel<<<1, 1024, 0, stream>>>(cand, reg[0], reg[1], reg[2], (float*)d_out);
}